// NGCF_51041391345797
// MI455X (gfx1250) — compile-verified
//
#include <hip/hip_runtime.h>
#include <hip/hip_bf16.h>

#define N_USER   100000
#define N_ITEM   50000
#define N_NODES  150000
#define NNZ      4800000
#define EMB      64
#define BATCH    16384
#define NEG_SLOPE 0.2f
#define EPS_NRM  1e-12f

typedef __attribute__((ext_vector_type(2))) float v2f;
typedef __attribute__((ext_vector_type(8))) float v8f;

// ---------------------------------------------------------------------------
// ego = concat(user_emb, item_emb); all_emb[:, 0:64] = ego  (layer-0, unnormalized)
// ---------------------------------------------------------------------------
__global__ __launch_bounds__(256)
void ngcf_init(const float* __restrict__ user_emb,
               const float* __restrict__ item_emb,
               float* __restrict__ ego,
               float* __restrict__ all_emb)
{
    long long t = (long long)blockIdx.x * 256 + threadIdx.x;  // over N_NODES*16 float4
    if (t >= (long long)N_NODES * 16) return;
    long long node = t >> 4;
    int q = (int)(t & 15);
    const float4* src = (node < N_USER)
        ? (const float4*)(user_emb + node * EMB)
        : (const float4*)(item_emb + (node - N_USER) * EMB);
    float4 v = src[q];
    ((float4*)ego)[t] = v;
    ((float4*)all_emb)[node * 64 + q] = v;   // row stride 256 floats = 64 float4
}

// ---------------------------------------------------------------------------
// side[r] += val * ego[c]  — one wave32 per nonzero, float2 per lane,
// f32 atomic adds resolve in L2 (side = 38MB << 192MB L2).
// ---------------------------------------------------------------------------
__global__ __launch_bounds__(256)
void ngcf_spmm(const int* __restrict__ rows,
               const int* __restrict__ cols,
               const float* __restrict__ vals,
               const float* __restrict__ ego,
               float* __restrict__ side)
{
    long long t = (long long)blockIdx.x * 256 + threadIdx.x;
    long long e = t >> 5;
    int lane = threadIdx.x & 31;
    if (e >= (long long)NNZ) return;
    int r = rows[e];
    int c = cols[e];
    float v = vals[e];
    float2 m = ((const float2*)(ego + (size_t)c * EMB))[lane];
    float* dst = side + (size_t)r * EMB + lane * 2;
    atomicAdd(dst,     v * m.x);
    atomicAdd(dst + 1, v * m.y);
}

// ---------------------------------------------------------------------------
// Fused layer:  ego = leaky_relu(side@Wgc + bgc + (ego*side)@Wbi + bbi)
//               all_emb[:, col0:col0+64] = ego / max(||ego_row||, eps)
// One 16-row tile per wave; 8 waves per WG; both weight matrices in LDS.
// WMMA: V_WMMA_F32_16X16X4_F32, 16 K-steps x 4 N-tiles x 2 GEMMs = 128/wave.
// ---------------------------------------------------------------------------
__global__ __launch_bounds__(256)
void ngcf_layer(const float* __restrict__ side,
                float* __restrict__ ego,          // read (bi term) + overwritten
                float* __restrict__ all_emb,
                const float* __restrict__ Wgc,    // [64,64] row-major (K x N)
                const float* __restrict__ bgc,    // [64]
                const float* __restrict__ Wbi,
                const float* __restrict__ bbi,
                int out_col0)
{
    __shared__ float sWgc[64 * 64];
    __shared__ float sWbi[64 * 64];
    int tid = threadIdx.x;
    for (int i = tid; i < 64 * 64; i += 256) {
        sWgc[i] = Wgc[i];
        sWbi[i] = Wbi[i];
    }
    __syncthreads();

    int wave = tid >> 5;
    int lane = tid & 31;
    int tileIdx = blockIdx.x * 8 + wave;
    if (tileIdx >= N_NODES / 16) return;           // wave-uniform: EXEC all-1 at WMMA

    int m0   = tileIdx * 16;
    int half = lane >> 4;     // 0: K={0,1}, rows 0-7 of C ; 1: K={2,3}, rows 8-15
    int l16  = lane & 15;
    int arow = m0 + l16;      // A-fragment row for this lane

    // C init = bgc[n] + bbi[n]  (N = l16 within each 16-col tile)
    v8f acc[4];
    #pragma unroll
    for (int nt = 0; nt < 4; ++nt) {
        float bias = bgc[nt * 16 + l16] + bbi[nt * 16 + l16];
        #pragma unroll
        for (int v = 0; v < 8; ++v) acc[nt][v] = bias;
    }

    const float* srow = side + (size_t)arow * EMB;
    const float* erow = ego  + (size_t)arow * EMB;

    #pragma unroll 4
    for (int kk = 0; kk < 16; ++kk) {
        int kbase = kk * 4 + half * 2;
        float sx = srow[kbase], sy = srow[kbase + 1];
        float ex = erow[kbase], ey = erow[kbase + 1];
        v2f a1; a1.x = sx;      a1.y = sy;        // A = side
        v2f a2; a2.x = ex * sx; a2.y = ey * sy;   // A = ego .* side
        #pragma unroll
        for (int nt = 0; nt < 4; ++nt) {
            int col = nt * 16 + l16;
            v2f bg, bb;
            bg.x = sWgc[kbase * 64 + col];        // B vgpr0: K = kbase
            bg.y = sWgc[(kbase + 1) * 64 + col];  // B vgpr1: K = kbase+1
            bb.x = sWbi[kbase * 64 + col];
            bb.y = sWbi[(kbase + 1) * 64 + col];
            acc[nt] = __builtin_amdgcn_wmma_f32_16x16x4_f32(
                false, a1, false, bg, (short)0, acc[nt], false, false);
            acc[nt] = __builtin_amdgcn_wmma_f32_16x16x4_f32(
                false, a2, false, bb, (short)0, acc[nt], false, false);
        }
    }

    // leaky-ReLU + per-row sum of squares (row = v + 8*half, partial over 4 cols)
    float part[8];
    #pragma unroll
    for (int v = 0; v < 8; ++v) {
        float s = 0.f;
        #pragma unroll
        for (int nt = 0; nt < 4; ++nt) {
            float x = acc[nt][v];
            x = (x > 0.f) ? x : NEG_SLOPE * x;
            acc[nt][v] = x;
            s += x * x;
        }
        part[v] = s;
    }
    // reduce across the 16 lanes of each half (masks 1,2,4,8 keep bit4 fixed)
    #pragma unroll
    for (int m = 1; m < 16; m <<= 1) {
        #pragma unroll
        for (int v = 0; v < 8; ++v) part[v] += __shfl_xor(part[v], m, 32);
    }
    float inv[8];
    #pragma unroll
    for (int v = 0; v < 8; ++v) {
        float n = fmaxf(sqrtf(part[v]), EPS_NRM);
        inv[v] = 1.0f / n;
    }

    // store: ego (unnormalized, feeds next layer) + all_emb slice (normalized)
    #pragma unroll
    for (int v = 0; v < 8; ++v) {
        int r = m0 + v + half * 8;
        #pragma unroll
        for (int nt = 0; nt < 4; ++nt) {
            int col = nt * 16 + l16;
            float x = acc[nt][v];
            ego[(size_t)r * EMB + col] = x;
            all_emb[(size_t)r * 256 + out_col0 + col] = x * inv[v];
        }
    }
}

// ---------------------------------------------------------------------------
// Head: out[b] = sigmoid( sum_j u[j]*i[j]*W_out[j] + b_out ), one wave per pair
// ---------------------------------------------------------------------------
__global__ __launch_bounds__(256)
void ngcf_head(const float* __restrict__ all_emb,
               const int* __restrict__ user_idx,
               const int* __restrict__ item_idx,
               const float* __restrict__ W_out,  // [256]
               const float* __restrict__ b_out,  // [1]
               float* __restrict__ out)
{
    int warp = (blockIdx.x * 256 + threadIdx.x) >> 5;
    int lane = threadIdx.x & 31;
    if (warp >= BATCH) return;
    const float* u = all_emb + (size_t)user_idx[warp] * 256;
    const float* it = all_emb + ((size_t)N_USER + item_idx[warp]) * 256;
    float s = 0.f;
    #pragma unroll
    for (int j = 0; j < 8; ++j) {
        int c = j * 32 + lane;           // coalesced across lanes
        s += u[c] * it[c] * W_out[c];
    }
    #pragma unroll
    for (int m = 16; m >= 1; m >>= 1) s += __shfl_xor(s, m, 32);
    if (lane == 0)
        out[warp] = 1.0f / (1.0f + expf(-(s + b_out[0])));
}

// ---------------------------------------------------------------------------
extern "C" void kernel_launch(void* const* d_in, const int* in_sizes, int n_in,
                              void* d_out, int out_size, void* d_ws, size_t ws_size,
                              hipStream_t stream)
{
    const int*   user_idx = (const int*)  d_in[0];
    const int*   item_idx = (const int*)  d_in[1];
    const int*   adj_rows = (const int*)  d_in[2];
    const int*   adj_cols = (const int*)  d_in[3];
    const float* adj_vals = (const float*)d_in[4];
    const float* user_emb = (const float*)d_in[5];
    const float* item_emb = (const float*)d_in[6];
    const float* W_gc     = (const float*)d_in[7];   // [3,64,64]
    const float* b_gc     = (const float*)d_in[8];   // [3,1,64]
    const float* W_bi     = (const float*)d_in[9];
    const float* b_bi     = (const float*)d_in[10];
    const float* W_out    = (const float*)d_in[11];  // [256,1]
    const float* b_out    = (const float*)d_in[12];

    float* ego     = (float*)d_ws;                       //  9.6M floats
    float* side    = ego  + (size_t)N_NODES * EMB;       //  9.6M floats
    float* all_emb = side + (size_t)N_NODES * EMB;       // 38.4M floats

    // ego = concat(embs); all_emb[:,0:64] = ego
    ngcf_init<<<(N_NODES * 16 + 255) / 256, 256, 0, stream>>>(
        user_emb, item_emb, ego, all_emb);

    const int spmm_blocks = (int)(((long long)NNZ * 32 + 255) / 256);   // 600000
    const int gemm_blocks = (N_NODES / 16 + 7) / 8;                      // 1172

    for (int k = 0; k < 3; ++k) {
        hipMemsetAsync(side, 0, (size_t)N_NODES * EMB * sizeof(float), stream);
        ngcf_spmm<<<spmm_blocks, 256, 0, stream>>>(
            adj_rows, adj_cols, adj_vals, ego, side);
        ngcf_layer<<<gemm_blocks, 256, 0, stream>>>(
            side, ego, all_emb,
            W_gc + (size_t)k * 64 * 64, b_gc + (size_t)k * 64,
            W_bi + (size_t)k * 64 * 64, b_bi + (size_t)k * 64,
            64 * (k + 1));
    }

    ngcf_head<<<(BATCH * 32) / 256, 256, 0, stream>>>(
        all_emb, user_idx, item_idx, W_out, b_out, (float*)d_out);
}